// RPN_66176856097086
// MI455X (gfx1250) — compile-verified
//
#include <hip/hip_runtime.h>
#include <hip/hip_bf16.h>

// ---------------------------------------------------------------------------
// Greedy NMS for RPN sampling, one workgroup per image.
//  - scores sorted in LDS via bitonic sort on packed u64 keys
//    key = ((0xFFFFFFFF - score_bits) << 32) | idx  -> ascending sort gives
//    score-descending, index-ascending (matches stable argsort(-scores)).
//  - sequential survivor scan by lane 0 (monotonic head pointer, O(N) total),
//    parallel IoU suppression sweep by all 1024 threads (32 wave32 waves).
//  - box records fetched as single b128 loads (16B-aligned float4 records).
//  - output rows written as boxes are kept; tail rows zero-filled.
//  - CDNA5 paths: gfx1250 async global->LDS DMA for the score stage,
//    s_wait_asynccnt, LDS-resident sort/flags (ds ops + split wait counters).
// ---------------------------------------------------------------------------

#define N_BOXES     4096
#define NMS_THREADS 1024

typedef __attribute__((address_space(1))) int* gint_ptr;   // global (AS1)
typedef __attribute__((address_space(3))) int* lint_ptr;   // LDS    (AS3)

__global__ __launch_bounds__(NMS_THREADS)
void rpn_nms_kernel(const float* __restrict__ scores,
                    const float* __restrict__ boxes,
                    const float* __restrict__ anchors,
                    const int*   __restrict__ nsamp_p,
                    float*       __restrict__ out)
{
    constexpr int   N         = N_BOXES;
    constexpr float IOU_THR   = 0.35f;
    constexpr float SCORE_THR = 0.5f;

    __shared__ unsigned long long s_key[N];   // 32 KB: sorted (score,idx) keys
    __shared__ float              s_stage[N]; // 16 KB: raw scores staging
    __shared__ unsigned char      s_supp[N];  //  4 KB: suppression flags
    __shared__ int   s_sel;
    __shared__ int   s_head;
    __shared__ int   s_nkept;
    __shared__ float s_box[4];

    const int tid = threadIdx.x;
    const int T   = blockDim.x;
    const int img = blockIdx.x;
    const int nsamp = nsamp_p[0];

    const float*  sc  = scores  + (size_t)img * N;
    const float4* bx4 = (const float4*)(boxes   + (size_t)img * (N * 4)); // 16B records
    const float*  an  = anchors + (size_t)img * (N * 4);
    float*        op  = out     + (size_t)img * (size_t)nsamp * 7;

    // ---- stage scores into LDS: async global->LDS DMA path on gfx1250 ----
#if defined(__gfx1250__) && __has_builtin(__builtin_amdgcn_global_load_async_to_lds_b32)
    for (int i = tid; i < N; i += T) {
        __builtin_amdgcn_global_load_async_to_lds_b32(
            (gint_ptr)(sc + i), (lint_ptr)(s_stage + i), 0, 0);
    }
#if __has_builtin(__builtin_amdgcn_s_wait_asynccnt)
    __builtin_amdgcn_s_wait_asynccnt(0);
#else
    asm volatile("s_wait_asynccnt 0" ::: "memory");
#endif
#else
    for (int i = tid; i < N; i += T) s_stage[i] = sc[i];
#endif
    __syncthreads();

    // ---- build sort keys + clear suppression flags ----
    for (int i = tid; i < N; i += T) {
        const unsigned int sb = __float_as_uint(s_stage[i]);  // scores in [0,1): positive
        s_key[i] = ((unsigned long long)(0xFFFFFFFFu - sb) << 32) | (unsigned int)i;
        s_supp[i] = 0;
    }
    if (tid == 0) { s_head = 0; s_nkept = 0; }
    __syncthreads();

    // ---- bitonic sort (ascending) over 4096 u64 keys in LDS ----
    for (int k = 2; k <= N; k <<= 1) {
        for (int j = k >> 1; j > 0; j >>= 1) {
            for (int i = tid; i < N; i += T) {
                const int ixj = i ^ j;
                if (ixj > i) {
                    const unsigned long long a = s_key[i];
                    const unsigned long long b = s_key[ixj];
                    const bool up = ((i & k) == 0);
                    if (up ? (a > b) : (a < b)) { s_key[i] = b; s_key[ixj] = a; }
                }
            }
            __syncthreads();
        }
    }

    // ---- greedy NMS ----
    for (;;) {
        if (tid == 0) {
            int sel = -1;
            if (s_nkept < nsamp) {
                int h = s_head;
                while (h < N) {
                    const unsigned long long kv = s_key[h];
                    const float scv =
                        __uint_as_float(0xFFFFFFFFu - (unsigned int)(kv >> 32));
                    if (!(scv > SCORE_THR)) break;   // sorted: rest invalid
                    if (!s_supp[h]) {
                        const int bi = (int)(unsigned int)kv;
                        const float4 b = bx4[bi];                 // one b128 load
                        s_box[0] = b.x; s_box[1] = b.y;
                        s_box[2] = b.z; s_box[3] = b.w;
                        const float2 aw =
                            *(const float2*)(an + (size_t)bi * 4 + 2); // b64 load
                        float* row = op + (size_t)s_nkept * 7;
                        row[0] = scv;
                        row[1] = b.x; row[2] = b.y; row[3] = b.z; row[4] = b.w;
                        row[5] = aw.x;
                        row[6] = aw.y;
                        s_nkept = s_nkept + 1;
                        sel = h;
                        break;
                    }
                    ++h;
                }
                s_head = (sel >= 0) ? (sel + 1) : N;
            }
            s_sel = sel;
        }
        __syncthreads();
        const int sel = s_sel;
        if (sel < 0) break;

        // parallel suppression sweep over remaining candidates
        const float ky1 = s_box[0], kx1 = s_box[1];
        const float ky2 = s_box[2], kx2 = s_box[3];
        const float karea = (ky2 - ky1) * (kx2 - kx1);
        for (int jj = sel + 1 + tid; jj < N; jj += T) {
            if (s_supp[jj]) continue;
            const int bi = (int)(unsigned int)s_key[jj];
            const float4 b = bx4[bi];                             // one b128 load
            const float iy1 = fmaxf(b.x, ky1);
            const float ix1 = fmaxf(b.y, kx1);
            const float iy2 = fminf(b.z, ky2);
            const float ix2 = fminf(b.w, kx2);
            const float inter = fmaxf(iy2 - iy1, 0.0f) * fmaxf(ix2 - ix1, 0.0f);
            const float uni   = karea + (b.z - b.x) * (b.w - b.y) - inter;
            const float iou   = (uni > 0.0f) ? (inter / uni) : 0.0f;
            if (iou > IOU_THR) s_supp[jj] = 1;
        }
        __syncthreads();   // supp flags visible to lane 0's next scan
    }

    // ---- zero-fill unused tail rows (d_out is poisoned by the harness) ----
    const int nk  = s_nkept;
    const int rem = (nsamp - nk) * 7;
    float* zp = op + (size_t)nk * 7;
    for (int p = tid; p < rem; p += T) zp[p] = 0.0f;
}

extern "C" void kernel_launch(void* const* d_in, const int* in_sizes, int n_in,
                              void* d_out, int out_size, void* d_ws, size_t ws_size,
                              hipStream_t stream) {
    const float* scores  = (const float*)d_in[0];   // (B, 64, 64, 1)
    const float* boxes   = (const float*)d_in[1];   // (B, 64, 64, 4)
    const float* anchors = (const float*)d_in[2];   // (B, 64, 64, 4)
    const int*   nsamp   = (const int*)d_in[3];     // scalar num_samples
    float*       out     = (float*)d_out;           // (B, num_samples, 7)

    const int B = in_sizes[0] / N_BOXES;            // boxes per image fixed at 4096
    if (B <= 0) return;

    rpn_nms_kernel<<<dim3(B), dim3(NMS_THREADS), 0, stream>>>(
        scores, boxes, anchors, nsamp, out);
}